// SAGEConv_85126251806777
// MI455X (gfx1250) — compile-verified
//
#include <hip/hip_runtime.h>
#include <math.h>

// ---------------------------------------------------------------------------
// SAGEConv forward for MI455X (gfx1250, wave32, WMMA)
//   xw   = x @ W                       (f32 WMMA 16x16x4, LDS-staged weight)
//   acc  = scatter_add(xw[col] -> row) + xw   (wave-per-edge f32 atomics, L2-resident)
//   out  = normalize(acc / (deg+1))    (wave-per-node shfl_xor reduction)
// ---------------------------------------------------------------------------

typedef __attribute__((ext_vector_type(2))) float v2f;
typedef __attribute__((ext_vector_type(8))) float v8f;

#define NN 100000   // nodes (multiple of 16)
#define DD 128      // feature dim (in == out)
#define EE 1600000  // edges

// ---- Kernel 1: xw = x @ W, also seed accumulator (d_out) with xw ----------
__global__ __launch_bounds__(256) void gemm_xw_kernel(const float* __restrict__ x,
                                                      const float* __restrict__ w,
                                                      float* __restrict__ xw,
                                                      float* __restrict__ acc) {
    __shared__ float s_w[DD * DD];  // 64 KB: full 128x128 f32 weight

    const int tid = threadIdx.x;
    // Cooperative weight load: 4096 float4 / 256 threads = 16 each
    {
        const float4* src = (const float4*)w;
        float4* dst = (float4*)s_w;
        #pragma unroll
        for (int i = 0; i < (DD * DD / 4) / 256; ++i)
            dst[tid + i * 256] = src[tid + i * 256];
    }
    __syncthreads();

    const int wave  = tid >> 5;          // 0..7 -> output col tile
    const int lane  = tid & 31;
    const int m0    = blockIdx.x * 16;   // row tile base (N % 16 == 0)
    const int c0    = wave * 16;         // col tile base
    const int mrow  = lane & 15;         // A row within tile
    const int ncol  = lane & 15;         // B/D col within tile
    const int khalf = (lane >> 4) << 1;  // 0 (lanes 0-15) or 2 (lanes 16-31)

    const float* arow = x + (size_t)(m0 + mrow) * DD;

    v8f accv = {};
    #pragma unroll
    for (int k0 = 0; k0 < DD; k0 += 4) {
        // 32-bit A 16x4 fragment: lane<16 -> K={k0,k0+1}; lane>=16 -> K={k0+2,k0+3}
        v2f a;
        a.x = arow[k0 + khalf];
        a.y = arow[k0 + khalf + 1];
        // B 4x16 fragment from LDS (mirrored layout); conflict-free banks
        v2f b;
        b.x = s_w[(k0 + khalf) * DD + c0 + ncol];
        b.y = s_w[(k0 + khalf + 1) * DD + c0 + ncol];
        accv = __builtin_amdgcn_wmma_f32_16x16x4_f32(
            /*neg_a=*/false, a, /*neg_b=*/false, b,
            /*c_mod=*/(short)0, accv, /*reuse_a=*/false, /*reuse_b=*/false);
    }

    // D layout: VGPR i -> row (m0 + i) for lanes 0-15, (m0 + 8 + i) for lanes 16-31
    const int mbase = m0 + ((lane >> 4) << 3);
    #pragma unroll
    for (int i = 0; i < 8; ++i) {
        const float v = accv[i];
        const size_t off = (size_t)(mbase + i) * DD + c0 + ncol;
        xw[off]  = v;  // clean copy for the edge gather
        acc[off] = v;  // self term of the scatter-mean (seed d_out)
    }
}

// ---- Kernel 2: degree counter init (cnt = 1.0, the self loop) -------------
__global__ void init_cnt_kernel(float* __restrict__ cnt) {
    const int i = blockIdx.x * blockDim.x + threadIdx.x;
    if (i < NN) cnt[i] = 1.0f;
}

// ---- Kernel 3: wave-per-edge scatter-add (xw and acc are L2-resident) -----
__global__ __launch_bounds__(256) void scatter_kernel(const int* __restrict__ row,
                                                      const int* __restrict__ col,
                                                      const float* __restrict__ xw,
                                                      float* __restrict__ acc,
                                                      float* __restrict__ cnt) {
    const int gtid = blockIdx.x * 256 + threadIdx.x;
    const int e    = gtid >> 5;
    const int lane = threadIdx.x & 31;
    if (e >= EE) return;
    const int r = row[e];
    const int c = col[e];
    if (r == c) return;  // remove_self_loops (uniform per wave)

    const float4 v = *(const float4*)(xw + (size_t)c * DD + lane * 4);
    float* dst = acc + (size_t)r * DD + lane * 4;
    atomicAdd(dst + 0, v.x);
    atomicAdd(dst + 1, v.y);
    atomicAdd(dst + 2, v.z);
    atomicAdd(dst + 3, v.w);
    if (lane == 0) atomicAdd(cnt + r, 1.0f);
}

// ---- Kernel 4: out = normalize(acc / cnt), wave-per-node ------------------
__global__ __launch_bounds__(256) void finalize_kernel(float* __restrict__ out,
                                                       const float* __restrict__ cnt) {
    const int gtid = blockIdx.x * 256 + threadIdx.x;
    const int node = gtid >> 5;
    const int lane = threadIdx.x & 31;
    if (node >= NN) return;

    float4 v = *(const float4*)(out + (size_t)node * DD + lane * 4);
    const float inv_c = 1.0f / cnt[node];
    v.x *= inv_c; v.y *= inv_c; v.z *= inv_c; v.w *= inv_c;

    float ss = v.x * v.x + v.y * v.y + v.z * v.z + v.w * v.w;
    #pragma unroll
    for (int off = 16; off > 0; off >>= 1)
        ss += __shfl_xor(ss, off, 32);   // wave32 reduction

    const float inv_n = 1.0f / fmaxf(sqrtf(ss), 1e-12f);
    v.x *= inv_n; v.y *= inv_n; v.z *= inv_n; v.w *= inv_n;
    *(float4*)(out + (size_t)node * DD + lane * 4) = v;
}

// ---------------------------------------------------------------------------
extern "C" void kernel_launch(void* const* d_in, const int* in_sizes, int n_in,
                              void* d_out, int out_size, void* d_ws, size_t ws_size,
                              hipStream_t stream) {
    (void)in_sizes; (void)n_in; (void)out_size; (void)ws_size;

    const float* x          = (const float*)d_in[0];
    const int*   edge_index = (const int*)d_in[1];   // [2, E] flat: row then col
    const float* weight     = (const float*)d_in[2];

    float* out = (float*)d_out;                      // doubles as scatter accumulator
    float* xw  = (float*)d_ws;                       // N*D f32
    float* cnt = xw + (size_t)NN * DD;               // N f32

    const int* row = edge_index;
    const int* col = edge_index + EE;

    init_cnt_kernel<<<(NN + 255) / 256, 256, 0, stream>>>(cnt);
    gemm_xw_kernel<<<NN / 16, 256, 0, stream>>>(x, weight, xw, out);
    scatter_kernel<<<(EE * 32 + 255) / 256, 256, 0, stream>>>(row, col, xw, out, cnt);
    finalize_kernel<<<(NN * 32 + 255) / 256, 256, 0, stream>>>(out, cnt);
}